// PFDetLoss_49469433315734
// MI455X (gfx1250) — compile-verified
//
#include <hip/hip_runtime.h>
#include <math.h>

// ---------------- constants from the reference ----------------
#define NT       256      // threads per block (8 wave32 waves)
#define MAXE     400      // max positive cells per image: 40 gts * 10 cands
#define NGT      40
#define BATCH    64
#define NUMCAND  10
#define EPSF     1e-7f
#define ALPHAF   0.25f
#define RADIUSF  2.5f
#define IMGF     1280.0f
#define OBJ_WF   1.0f
#define BOX_WF   5.0f
#define FOOT_WF  1.0f

// ---------------- small device helpers ----------------
__device__ __forceinline__ float sigm(float x) { return 1.0f / (1.0f + expf(-x)); }
__device__ __forceinline__ float softplusf(float x) { return fmaxf(x, 0.0f) + log1pf(expf(-fabsf(x))); }

// Monotonic float->uint mapping packed with index; min() gives (min value, then min index).
__device__ __forceinline__ unsigned long long packKey(float v, int i) {
    unsigned int u = __float_as_uint(v);
    u = (u & 0x80000000u) ? ~u : (u | 0x80000000u);
    return (((unsigned long long)u) << 32) | (unsigned int)i;
}

__device__ __forceinline__ float iouXYXY(float ax1, float ay1, float ax2, float ay2,
                                         float bx1, float by1, float bx2, float by2,
                                         float aArea, float bArea) {
    float iw = fmaxf(fminf(ax2, bx2) - fmaxf(ax1, bx1), 0.0f);
    float ih = fmaxf(fminf(ay2, by2) - fmaxf(ay1, by1), 0.0f);
    float inter = iw * ih;
    return inter / (aArea + bArea - inter + EPSF);
}

__device__ __forceinline__ float ciouF(float pcx, float pcy, float pw, float ph,
                                       float tcx, float tcy, float tw, float th) {
    float px1 = pcx - 0.5f * pw, py1 = pcy - 0.5f * ph, px2 = pcx + 0.5f * pw, py2 = pcy + 0.5f * ph;
    float tx1 = tcx - 0.5f * tw, ty1 = tcy - 0.5f * th, tx2 = tcx + 0.5f * tw, ty2 = tcy + 0.5f * th;
    float iw = fmaxf(fminf(px2, tx2) - fmaxf(px1, tx1), 0.0f);
    float ih = fmaxf(fminf(py2, ty2) - fmaxf(py1, ty1), 0.0f);
    float inter = iw * ih;
    float uni = pw * ph + tw * th - inter + EPSF;
    float iou = inter / uni;
    float dx = pcx - tcx, dy = pcy - tcy;
    float rho2 = dx * dx + dy * dy;
    float cw = fmaxf(px2, tx2) - fminf(px1, tx1);
    float ch = fmaxf(py2, ty2) - fminf(py1, ty1);
    float c2 = cw * cw + ch * ch + EPSF;
    float da = atanf(tw / (th + EPSF)) - atanf(pw / (ph + EPSF));
    float v = 0.40528473456935108577f * da * da;  // 4/pi^2
    float alpha = v / (1.0f - iou + v + EPSF);
    return iou - rho2 / c2 - alpha * v;
}

// decode one cell (pcx,pcy,pw,ph in normalized units; cls_cost = softplus(-p0))
__device__ __forceinline__ void decodeCell(const float* __restrict__ pb, int HW, int W, int i, float s,
                                           float& pcx, float& pcy, float& pw, float& ph, float& clsc) {
    float gx = (float)(i % W);
    float gy = (float)(i / W);
    float x0 = pb[i];
    pcx = (sigm(pb[HW + i]) + gx) * s;
    pcy = (sigm(pb[2 * HW + i]) + gy) * s;
    float w3 = fminf(fmaxf(pb[3 * HW + i], -5.0f), 5.0f);
    float w4 = fminf(fmaxf(pb[4 * HW + i], -5.0f), 5.0f);
    pw = expf(w3) * s;
    ph = expf(w4) * s;
    clsc = fmaxf(-x0, 0.0f) + log1pf(expf(-fabsf(x0)));
}

// ---------------- block reductions (blockDim.x == NT, fixed order => deterministic) ----------------
__device__ __forceinline__ float blockSumF(float v, float* sc) {
    int t = threadIdx.x;
    sc[t] = v; __syncthreads();
    for (int o = NT / 2; o > 0; o >>= 1) { if (t < o) sc[t] += sc[t + o]; __syncthreads(); }
    float r = sc[0]; __syncthreads();
    return r;
}
__device__ __forceinline__ int blockSumI(int v, int* sc) {
    int t = threadIdx.x;
    sc[t] = v; __syncthreads();
    for (int o = NT / 2; o > 0; o >>= 1) { if (t < o) sc[t] += sc[t + o]; __syncthreads(); }
    int r = sc[0]; __syncthreads();
    return r;
}
__device__ __forceinline__ unsigned long long blockMinU64(unsigned long long v, unsigned long long* sc) {
    int t = threadIdx.x;
    sc[t] = v; __syncthreads();
    for (int o = NT / 2; o > 0; o >>= 1) {
        if (t < o) { unsigned long long b = sc[t + o]; if (b < sc[t]) sc[t] = b; }
        __syncthreads();
    }
    unsigned long long r = sc[0]; __syncthreads();
    return r;
}

// ---------------- main fused assignment + loss kernel: one block per (image, level) ----------------
template <int LEVEL, int H, int W, int STRIDE>
__global__ __launch_bounds__(NT)
void pfdetLoss(const float* __restrict__ pred, const float* __restrict__ targets,
               float* __restrict__ decw, float* __restrict__ partials) {
    constexpr int HW = H * W;
    extern __shared__ unsigned char smem[];
    float* costL = (float*)smem;             // HW floats: per-gt assignment cost
    short* mapL  = (short*)(costL + HW);     // HW shorts: cell -> table entry (-1 = none)

    __shared__ float sf[NT];
    __shared__ int   si[NT];
    __shared__ unsigned long long su[NT];
    __shared__ int   selIdx[NUMCAND];
    __shared__ int   s_count, s_ncand;
    __shared__ float s_iousum, s_bestiou;
    __shared__ int   cellE[MAXE];
    __shared__ float objE[MAXE];
    __shared__ float b0E[MAXE], b1E[MAXE], b2E[MAXE], b3E[MAXE], f0E[MAXE], f1E[MAXE];

    const int b = blockIdx.x;
    const int tid = threadIdx.x;
    const float s = (float)STRIDE / IMGF;
    const float inv_s = 1.0f / s;
    const float* pb = pred + (size_t)b * 7 * HW;
    float* dec = decw + (size_t)b * 5 * HW;   // L2-resident decode cache (own-cell access only)

    if (tid == 0) s_count = 0;
    for (int i = tid; i < HW; i += NT) mapL[i] = -1;

    // ---- Pass A: decode each cell once; accumulate baseline focal term for z = 0 ----
    float obj0 = 0.0f;
    for (int i = tid; i < HW; i += NT) {
        float pcx, pcy, pw, ph, clsc;
        decodeCell(pb, HW, W, i, s, pcx, pcy, pw, ph, clsc);
        dec[i] = pcx; dec[HW + i] = pcy; dec[2 * HW + i] = pw; dec[3 * HW + i] = ph; dec[4 * HW + i] = clsc;
        float x = pb[i];
        float prob = sigm(x);
        obj0 += (1.0f - ALPHAF) * prob * prob * softplusf(x);
    }
    __syncthreads();

    // ---- Sequential GT scan (preserves last-writer-wins .set semantics) ----
    for (int g = 0; g < NGT; ++g) {
        if (g + 1 < NGT)  // CDNA5 global_prefetch_b8 of next gt row
            __builtin_prefetch(targets + ((size_t)b * NGT + g + 1) * 7, 0, 0);

        const float* tg = targets + ((size_t)b * NGT + g) * 7;
        float cls = tg[0];
        float cx = tg[1], cy = tg[2], w = tg[3], h = tg[4], fx = tg[5], fy = tg[6];
        float szp = fmaxf(w, h) * IMGF;
        bool smask = (LEVEL == 0) ? (szp < 128.0f)
                   : (LEVEL == 1) ? (szp >= 48.0f && szp < 288.0f)
                                  : (szp >= 128.0f);
        if (!((cls == 0.0f) && smask)) continue;  // invalid gt is a provable no-op in the scan

        float cxg = cx * inv_s, cyg = cy * inv_s;
        float bx1 = cx - 0.5f * w, by1 = cy - 0.5f * h, bx2 = cx + 0.5f * w, by2 = cy + 0.5f * h;
        float lox = bx1 * inv_s, hix = bx2 * inv_s, loy = by1 * inv_s, hiy = by2 * inv_s;
        float gtArea = w * h;

        // per-cell candidate / iou / cost pass (each thread touches only cells it decoded)
        int ncand = 0; float iousum = 0.0f;
        unsigned long long dmin = ~0ull;
        for (int i = tid; i < HW; i += NT) {
            float gx = (float)(i % W), gy = (float)(i / W);
            float dx = gx - cxg, dy = gy - cyg;
            bool inc = (fabsf(dx) < RADIUSF) && (fabsf(dy) < RADIUSF);
            bool inb = (gx >= lox) && (gx < hix) && (gy >= loy) && (gy < hiy);
            float dist = dx * dx + dy * dy;
            unsigned long long k = packKey(dist, i);
            if (k < dmin) dmin = k;
            float cost = __builtin_inff();
            if (inc || inb) {
                float pcx = dec[i], pcy = dec[HW + i], pw = dec[2 * HW + i], ph = dec[3 * HW + i];
                float iou = iouXYXY(pcx - 0.5f * pw, pcy - 0.5f * ph, pcx + 0.5f * pw, pcy + 0.5f * ph,
                                    bx1, by1, bx2, by2, pw * ph, gtArea);
                cost = dec[4 * HW + i] - 3.0f * logf(iou + EPSF);
                ncand++; iousum += iou;
            }
            costL[i] = cost;
        }
        int ncR = blockSumI(ncand, si);
        float iosR = blockSumF(iousum, sf);
        unsigned long long dR = blockMinU64(dmin, su);

        if (tid == 0) {
            if (ncR == 0) {  // fallback: nearest cell becomes the only candidate
                int c = (int)(dR & 0xffffffffu);
                float pcx, pcy, pw, ph, clsc;
                decodeCell(pb, HW, W, c, s, pcx, pcy, pw, ph, clsc);
                float iou = iouXYXY(pcx - 0.5f * pw, pcy - 0.5f * ph, pcx + 0.5f * pw, pcy + 0.5f * ph,
                                    bx1, by1, bx2, by2, pw * ph, gtArea);
                costL[c] = clsc - 3.0f * logf(iou + EPSF);
                s_ncand = 1; s_iousum = iou;
            } else { s_ncand = ncR; s_iousum = iosR; }
        }
        __syncthreads();

        int nCand = s_ncand;
        int npMax = nCand < NUMCAND ? nCand : NUMCAND;
        int npos_ = (int)floorf(s_iousum);
        if (npos_ < 1) npos_ = 1;
        if (npos_ > npMax) npos_ = npMax;

        // extract npos_ smallest costs (top_k of -cost; ties -> lower index, via packKey)
        for (int j = 0; j < npos_; ++j) {
            unsigned long long kmin = ~0ull;
            for (int i = tid; i < HW; i += NT) {
                unsigned long long k = packKey(costL[i], i);
                if (k < kmin) kmin = k;
            }
            unsigned long long r = blockMinU64(kmin, su);
            if (tid == 0) {
                int c = (int)(r & 0xffffffffu);
                selIdx[j] = c;
                costL[c] = __builtin_inff();
                if (j == 0) {  // best_iou = iou at best-cost cell
                    float pcx, pcy, pw, ph, clsc;
                    decodeCell(pb, HW, W, c, s, pcx, pcy, pw, ph, clsc);
                    s_bestiou = iouXYXY(pcx - 0.5f * pw, pcy - 0.5f * ph, pcx + 0.5f * pw, pcy + 0.5f * ph,
                                        bx1, by1, bx2, by2, pw * ph, gtArea);
                }
            }
            __syncthreads();
        }

        // serial (thread 0) table update => deterministic entry order; <= 10 entries per gt
        if (tid == 0) {
            float twt = logf(w * inv_s + EPSF);
            float tht = logf(h * inv_s + EPSF);
            for (int j = 0; j < npos_; ++j) {
                int c = selIdx[j];
                int e = mapL[c];
                if (e < 0) {
                    e = s_count++;
                    mapL[c] = (short)e;
                    cellE[e] = c;
                    objE[e] = 0.0f;
                }
                objE[e] = fmaxf(objE[e], s_bestiou);          // obj_t .max semantics
                float gx = (float)(c % W), gy = (float)(c / W);
                b0E[e] = cxg - gx; b1E[e] = cyg - gy;          // box_t .set (last gt wins)
                b2E[e] = twt;      b3E[e] = tht;
                f0E[e] = fx;       f1E[e] = fy;
            }
        }
        __syncthreads();
    }

    // ---- losses: baseline + corrections over positive cells only ----
    int cnt = s_count;
    float corr = 0.0f, boxs = 0.0f, foots = 0.0f;
    for (int e = tid; e < cnt; e += NT) {
        int c = cellE[e];
        float x = pb[c];
        float z = objE[e];
        float prob = sigm(x);
        float l1pe = log1pf(expf(-fabsf(x)));
        float ce = fmaxf(x, 0.0f) - x * z + l1pe;
        float pt = prob * z + (1.0f - prob) * (1.0f - z);
        float at = ALPHAF * z + (1.0f - ALPHAF) * (1.0f - z);
        float fz = at * (1.0f - pt) * (1.0f - pt) * ce;
        float f0v = (1.0f - ALPHAF) * prob * prob * (fmaxf(x, 0.0f) + l1pe);
        corr += fz - f0v;

        float pcx, pcy, pw, ph, clsc;
        decodeCell(pb, HW, W, c, s, pcx, pcy, pw, ph, clsc);  // recompute (no cross-thread ws reads)
        float gx = (float)(c % W), gy = (float)(c / W);
        float tcx = (b0E[e] + gx) * s, tcy = (b1E[e] + gy) * s;
        float tw = expf(b2E[e]) * s, th = expf(b3E[e]) * s;
        boxs += 1.0f - ciouF(pcx, pcy, pw, ph, tcx, tcy, tw, th);

        float p5 = sigm(pb[5 * HW + c]);
        float p6 = sigm(pb[6 * HW + c]);
        float d0 = fabsf(p5 - f0E[e]);
        float d1 = fabsf(p6 - f1E[e]);
        foots += (d0 < 1.0f ? 0.5f * d0 * d0 : d0 - 0.5f) + (d1 < 1.0f ? 0.5f * d1 * d1 : d1 - 0.5f);
    }

    float objR  = blockSumF(obj0, sf);
    float corrR = blockSumF(corr, sf);
    float boxR  = blockSumF(boxs, sf);
    float footR = blockSumF(foots, sf);
    if (tid == 0) {
        float nposf = fmaxf((float)cnt, 1.0f);
        float objLoss = (objR + corrR) / (float)HW;
        partials[b] = OBJ_WF * objLoss + BOX_WF * (boxR / nposf) + FOOT_WF * (footR / nposf);
    }
}

// ---------------- final reduction of 192 partials via V_WMMA_F32_16X16X4_F32 ----------------
typedef __attribute__((ext_vector_type(2))) float v2f;
typedef __attribute__((ext_vector_type(8))) float v8f;

__global__ void reducePartials(const float* __restrict__ partials, float* __restrict__ out) {
    int lane = threadIdx.x;  // exactly 32 threads, EXEC all ones, no divergence before WMMA
    v8f acc = {};
    v2f onesv; onesv[0] = 1.0f; onesv[1] = 1.0f;   // B = 4x16 all-ones (layout-agnostic)
    for (int base = 0; base < 3 * BATCH; base += 64) {
        v2f a;                                     // A = 16x4: 64 partials per WMMA
        a[0] = partials[base + 2 * lane];
        a[1] = partials[base + 2 * lane + 1];
        // D[m][n] += sum_k A[m][k]  (row sums replicated over n); chained C accumulation
        acc = __builtin_amdgcn_wmma_f32_16x16x4_f32(false, a, false, onesv, (short)0, acc, false, false);
    }
    float t = 0.0f;
#pragma unroll
    for (int j = 0; j < 8; ++j) t += acc[j];       // sum this lane's half-column of D
    t += __shfl_xor(t, 16, 32);                    // add other M-half: every lane now holds the total
    if (lane == 0) out[0] = t * (1.0f / (float)BATCH);
}

// ---------------- launcher ----------------
extern "C" void kernel_launch(void* const* d_in, const int* in_sizes, int n_in,
                              void* d_out, int out_size, void* d_ws, size_t ws_size,
                              hipStream_t stream) {
    (void)in_sizes; (void)n_in; (void)out_size; (void)ws_size;
    const float* p0 = (const float*)d_in[0];   // (64,7,160,160)
    const float* p1 = (const float*)d_in[1];   // (64,7,80,80)
    const float* p2 = (const float*)d_in[2];   // (64,7,40,40)
    const float* tg = (const float*)d_in[3];   // (64,40,7)

    float* wsf = (float*)d_ws;
    float* partials = wsf;                     // 192 floats (padded to 256)
    float* dec0 = wsf + 256;
    float* dec1 = dec0 + (size_t)BATCH * 5 * 160 * 160;
    float* dec2 = dec1 + (size_t)BATCH * 5 * 80 * 80;

    // dynamic LDS: cost[HW] floats + map[HW] shorts (level0: ~154 KB of the 320 KB WGP LDS)
    pfdetLoss<0, 160, 160, 8><<<BATCH, NT, 160 * 160 * 6 + 64, stream>>>(p0, tg, dec0, partials + 0 * BATCH);
    pfdetLoss<1, 80, 80, 16><<<BATCH, NT, 80 * 80 * 6 + 64, stream>>>(p1, tg, dec1, partials + 1 * BATCH);
    pfdetLoss<2, 40, 40, 32><<<BATCH, NT, 40 * 40 * 6 + 64, stream>>>(p2, tg, dec2, partials + 2 * BATCH);

    reducePartials<<<1, 32, 0, stream>>>(partials, (float*)d_out);
}